// WKV_21938692948060
// MI455X (gfx1250) — compile-verified
//
#include <hip/hip_runtime.h>
#include <stdint.h>

// RWKV WKV forward, chunked associative scan, CDNA5 (gfx1250).
// Memory-bound: ~671MB minimum traffic -> ~29us at 23.3 TB/s. WMMA is useless
// here (per-channel scalar decay); the MI455X win is the async Global->LDS
// path (ASYNCcnt) for deep, VGPR-free prefetch pipelines.

#define B_   8
#define T_   4096
#define C_   1024
#define L_   64            // chunk length (timesteps)
#define NC_  64            // T_/L_ chunks
#define TS_  8             // timesteps per staged LDS slice
#define NS_  (L_ / TS_)    // slices per chunk
#define CPB_ 256           // channels per block
#define EPS_ 1e-6f

// Copy a TS_ x CPB_ float tile (8KB) from global to LDS using async DMA.
// 512 x 16B transfers; each of the 8 waves issues 2 b128 instructions.
__device__ __forceinline__ void async_tile_load(const float* __restrict__ gsrc,
                                                const float* sdst, int tid) {
  uint32_t lbase = (uint32_t)(uintptr_t)(const void*)sdst;  // low 32 bits == LDS byte offset
#pragma unroll
  for (int j = 0; j < 2; ++j) {
    int idx = tid + (j << 8);            // 16-byte transfer id, 0..511
    int row = idx >> 6;                  // timestep within slice (0..7)
    int col = (idx & 63) << 2;           // float column (0..252)
    uint64_t ga = (uint64_t)(uintptr_t)(gsrc + (size_t)row * C_ + col);
    uint32_t la = lbase + (uint32_t)(idx << 4);
    asm volatile("global_load_async_to_lds_b128 %0, %1, off"
                 :: "v"(la), "v"(ga) : "memory");
  }
}

// ---------------- Pass 1: per-chunk affine summaries (D, A, B) ----------------
__global__ __launch_bounds__(CPB_) void wkv_pass1(
    const float* __restrict__ w, const float* __restrict__ k,
    const float* __restrict__ v,
    float* __restrict__ Dws, float* __restrict__ Aws, float* __restrict__ Bws) {
  __shared__ float sbuf[2][3][TS_ * CPB_];   // 48 KB, double buffered
  const int tid = threadIdx.x;
  const int blk = blockIdx.x;                // (b*NC_ + chunk)*4 + cb
  const int cb  = blk & 3;
  const int ch  = (blk >> 2) & (NC_ - 1);
  const int b   = blk >> 8;
  const int c0  = cb << 8;
  const int t0  = ch * L_;

  const size_t base = ((size_t)b * T_ + t0) * C_ + c0;
  async_tile_load(w + base, sbuf[0][0], tid);
  async_tile_load(k + base, sbuf[0][1], tid);
  async_tile_load(v + base, sbuf[0][2], tid);

  float D = 1.f, A = 0.f, Bc = 0.f;
#pragma unroll 1
  for (int s = 0; s < NS_; ++s) {
    if (s + 1 < NS_) {
      const size_t nb = base + (size_t)(s + 1) * TS_ * C_;
      const int nbuf = (s + 1) & 1;
      async_tile_load(w + nb, sbuf[nbuf][0], tid);
      async_tile_load(k + nb, sbuf[nbuf][1], tid);
      async_tile_load(v + nb, sbuf[nbuf][2], tid);
      asm volatile("s_wait_asynccnt 6" ::: "memory");   // slice s done (in-order)
    } else {
      asm volatile("s_wait_asynccnt 0" ::: "memory");
    }
    __syncthreads();
    const int cur = s & 1;
    const float* sw = sbuf[cur][0];
    const float* sk = sbuf[cur][1];
    const float* sv = sbuf[cur][2];
#pragma unroll
    for (int t = 0; t < TS_; ++t) {
      const float wv = sw[t * CPB_ + tid];
      const float kv = sk[t * CPB_ + tid];
      const float vv = sv[t * CPB_ + tid];
      const float d  = __expf(-__expf(wv));
      const float ek = __expf(kv);
      A  = fmaf(d, A, ek * vv);
      Bc = fmaf(d, Bc, ek);
      D *= d;
    }
    __syncthreads();   // buffer s&1 reused by slice s+2's async writes
  }
  const size_t o = ((size_t)b * NC_ + ch) * C_ + c0 + tid;
  Dws[o] = D; Aws[o] = A; Bws[o] = Bc;
}

// --------- Pass 2: scan chunk summaries; store incoming state in-place ---------
__global__ __launch_bounds__(CPB_) void wkv_pass2(
    const float* __restrict__ Dws, float* __restrict__ Aws, float* __restrict__ Bws) {
  const int g = blockIdx.x * CPB_ + threadIdx.x;   // 0 .. B_*C_-1
  const int c = g & (C_ - 1);
  const int b = g >> 10;
  float a = 0.f, bb = 0.f;
#pragma unroll 1
  for (int i = 0; i < NC_; ++i) {
    const size_t o = ((size_t)b * NC_ + i) * C_ + c;
    const float Di = Dws[o];
    const float Ai = Aws[o];
    const float Bi = Bws[o];
    Aws[o] = a;          // state entering chunk i
    Bws[o] = bb;
    a  = fmaf(Di, a, Ai);
    bb = fmaf(Di, bb, Bi);
  }
}

// ---------------- Pass 3: replay recurrence with seeds, emit output ----------------
__global__ __launch_bounds__(CPB_) void wkv_pass3(
    const float* __restrict__ r, const float* __restrict__ w,
    const float* __restrict__ k, const float* __restrict__ v,
    const float* __restrict__ Ain, const float* __restrict__ Bin,
    float* __restrict__ out) {
  __shared__ float sbuf[2][4][TS_ * CPB_];   // 64 KB, double buffered
  const int tid = threadIdx.x;
  const int blk = blockIdx.x;
  const int cb  = blk & 3;
  const int ch  = (blk >> 2) & (NC_ - 1);
  const int b   = blk >> 8;
  const int c0  = cb << 8;
  const int t0  = ch * L_;

  const size_t so = ((size_t)b * NC_ + ch) * C_ + c0 + tid;
  float a  = Ain[so];
  float bb = Bin[so];

  const size_t base = ((size_t)b * T_ + t0) * C_ + c0;
  async_tile_load(r + base, sbuf[0][0], tid);
  async_tile_load(w + base, sbuf[0][1], tid);
  async_tile_load(k + base, sbuf[0][2], tid);
  async_tile_load(v + base, sbuf[0][3], tid);

#pragma unroll 1
  for (int s = 0; s < NS_; ++s) {
    if (s + 1 < NS_) {
      const size_t nb = base + (size_t)(s + 1) * TS_ * C_;
      const int nbuf = (s + 1) & 1;
      async_tile_load(r + nb, sbuf[nbuf][0], tid);
      async_tile_load(w + nb, sbuf[nbuf][1], tid);
      async_tile_load(k + nb, sbuf[nbuf][2], tid);
      async_tile_load(v + nb, sbuf[nbuf][3], tid);
      asm volatile("s_wait_asynccnt 8" ::: "memory");
    } else {
      asm volatile("s_wait_asynccnt 0" ::: "memory");
    }
    __syncthreads();
    const int cur = s & 1;
    const float* sr = sbuf[cur][0];
    const float* sw = sbuf[cur][1];
    const float* sk = sbuf[cur][2];
    const float* sv = sbuf[cur][3];
#pragma unroll
    for (int t = 0; t < TS_; ++t) {
      const float rv = sr[t * CPB_ + tid];
      const float wv = sw[t * CPB_ + tid];
      const float kv = sk[t * CPB_ + tid];
      const float vv = sv[t * CPB_ + tid];
      const float d  = __expf(-__expf(wv));
      const float ek = __expf(kv);
      a  = fmaf(d, a, ek * vv);
      bb = fmaf(d, bb, ek);
      const float sig = 1.f / (1.f + __expf(-rv));
      const float o   = sig * __fdividef(a, bb + EPS_);
      out[((size_t)b * T_ + (t0 + s * TS_ + t)) * C_ + c0 + tid] = o;
    }
    __syncthreads();
  }
}

extern "C" void kernel_launch(void* const* d_in, const int* in_sizes, int n_in,
                              void* d_out, int out_size, void* d_ws, size_t ws_size,
                              hipStream_t stream) {
  (void)in_sizes; (void)n_in; (void)out_size; (void)ws_size;
  const float* r = (const float*)d_in[0];
  const float* w = (const float*)d_in[1];
  const float* k = (const float*)d_in[2];
  const float* v = (const float*)d_in[3];
  float* out = (float*)d_out;

  const size_t nsum = (size_t)B_ * NC_ * C_;   // 524288 floats per array
  float* Dws = (float*)d_ws;                   // 6 MB of d_ws used
  float* Aws = Dws + nsum;
  float* Bws = Aws + nsum;

  const int blocks = B_ * NC_ * (C_ / CPB_);   // 2048
  wkv_pass1<<<blocks, CPB_, 0, stream>>>(w, k, v, Dws, Aws, Bws);
  wkv_pass2<<<(B_ * C_) / CPB_, CPB_, 0, stream>>>(Dws, Aws, Bws);
  wkv_pass3<<<blocks, CPB_, 0, stream>>>(r, w, k, v, Aws, Bws, out);
}